// ChannelAttention_53300544143507
// MI455X (gfx1250) — compile-verified
//
#include <hip/hip_runtime.h>
#include <hip/hip_bf16.h>
#include <stdint.h>

typedef float v2f __attribute__((ext_vector_type(2)));
typedef float v8f __attribute__((ext_vector_type(8)));

#define BATCH 16
#define CH    256
#define TT    8192
#define EMB   32

static __device__ __forceinline__ v8f wmma_f32(v2f a, v2f b, v8f c) {
    // D = A(16x4 f32) * B(4x16 f32) + C(16x16 f32)
    return __builtin_amdgcn_wmma_f32_16x16x4_f32(
        /*neg_a=*/false, a, /*neg_b=*/false, b,
        /*c_mod=*/(short)0, c, /*reuse_a=*/false, /*reuse_b=*/false);
}

// Generic pointer to LDS keeps the LDS byte offset in addr[31:0] (flat->LDS
// aperture rule), so truncation yields the VGPR value ASYNC_TO_LDS expects.
static __device__ __forceinline__ uint32_t lds_off(const void* p) {
    return (uint32_t)(uintptr_t)p;
}

// 16B-per-lane async DMA: global memory -> LDS, tracked by ASYNCcnt.
static __device__ __forceinline__ void async_copy_b128(uint32_t lds_dst,
                                                       const void* gaddr) {
    asm volatile("global_load_async_to_lds_b128 %0, %1, off"
                 :
                 : "v"(lds_dst), "v"(gaddr)
                 : "memory");
}

static __device__ __forceinline__ void wait_async0() {
    asm volatile("s_wait_asynccnt 0" ::: "memory");
}

// ---------------------------------------------------------------------------
// Kernel 1: per-(b,t) channel mean + per-block partial of sum_c (x-mean)^2.
// ---------------------------------------------------------------------------
__global__ void k_mean_ssq(const float* __restrict__ x,
                           float* __restrict__ mean,
                           float* __restrict__ partials) {
    int g = blockIdx.x * 256 + threadIdx.x;
    int b = g >> 13;
    int t = g & (TT - 1);
    const float* xp = x + (size_t)b * CH * TT + t;
    float s = 0.0f, s2 = 0.0f;
#pragma unroll 8
    for (int c = 0; c < CH; ++c) {
        float v = xp[(size_t)c * TT];   // lanes -> consecutive t: coalesced
        s += v;
        s2 += v * v;
    }
    mean[g] = s * (1.0f / CH);
    float ssq = s2 - s * s * (1.0f / CH);

    __shared__ float red[256];
    red[threadIdx.x] = ssq;
    __syncthreads();
    for (int off = 128; off > 0; off >>= 1) {
        if (threadIdx.x < off) red[threadIdx.x] += red[threadIdx.x + off];
        __syncthreads();
    }
    if (threadIdx.x == 0) partials[blockIdx.x] = red[0];
}

// ---------------------------------------------------------------------------
// Kernel 2: u[b,c] = sum_t x[b,c,t]*mean[b,t]  (one block per (b,c) row).
// ---------------------------------------------------------------------------
__global__ void k_u(const float* __restrict__ x,
                    const float* __restrict__ mean,
                    float* __restrict__ u) {
    int bc = blockIdx.x;
    int b  = bc >> 8;
    const float* xp = x + (size_t)bc * TT;
    const float* mb = mean + (size_t)b * TT;
    float s = 0.0f;
    for (int t = threadIdx.x; t < TT; t += 256) s += xp[t] * mb[t];
    __shared__ float red[256];
    red[threadIdx.x] = s;
    __syncthreads();
    for (int off = 128; off > 0; off >>= 1) {
        if (threadIdx.x < off) red[threadIdx.x] += red[threadIdx.x + off];
        __syncthreads();
    }
    if (threadIdx.x == 0) u[bc] = red[0];
}

// ---------------------------------------------------------------------------
// Kernel 3: scale[b] = 255 / sum(ct^2)   and   s2[b] = sum_t mean^2.
// ---------------------------------------------------------------------------
__global__ void k_scale(const float* __restrict__ partials,
                        const float* __restrict__ mean,
                        float* __restrict__ scale,
                        float* __restrict__ s2) {
    int b = blockIdx.x, tid = threadIdx.x;
    float sp = (tid < 32) ? partials[b * 32 + tid] : 0.0f;
    float sm = 0.0f;
    const float* mb = mean + (size_t)b * TT;
    for (int t = tid; t < TT; t += 256) { float m = mb[t]; sm += m * m; }
    __shared__ float r1[256], r2[256];
    r1[tid] = sp; r2[tid] = sm;
    __syncthreads();
    for (int off = 128; off > 0; off >>= 1) {
        if (tid < off) { r1[tid] += r1[tid + off]; r2[tid] += r2[tid + off]; }
        __syncthreads();
    }
    if (tid == 0) { scale[b] = 255.0f / r1[0]; s2[b] = r2[0]; }
}

// ---------------------------------------------------------------------------
// Kernel 4: Gram matrix G = x@x^T via WMMA with async-DMA double buffering;
// epilogue applies pcc = (G - u_m - u_n + s2)*scale  (== centered cov / denom).
// Block = 256 threads (8 waves), 64x64 output tile; K chunks of 32 t-values.
// ---------------------------------------------------------------------------
__global__ void k_cov(const float* __restrict__ x,
                      const float* __restrict__ u,
                      const float* __restrict__ scale,
                      const float* __restrict__ s2,
                      float* __restrict__ pcc) {
    int b  = blockIdx.x;
    int m0 = blockIdx.y * 64;
    int n0 = blockIdx.z * 64;

    __shared__ float lA[2][64][36];   // 144B row stride: 16B-aligned, conflict-free
    __shared__ float lB[2][64][36];

    int tid  = threadIdx.x;
    int lane = tid & 31;
    int w    = tid >> 5;
    int wm   = w & 3;
    int wn   = w >> 2;

    const float* xb = x + (size_t)b * CH * TT;

    // each thread DMAs 2x16B per matrix per chunk (64 rows x 32 cols)
    int r0 = (tid * 4) >> 5, c0 = (tid * 4) & 31;          // j = 0
    int r1 = (tid * 4 + 1024) >> 5, c1 = (tid * 4 + 1024) & 31;  // j = 1

    v8f acc0 = {};
    v8f acc1 = {};

    // prologue: chunk 0 -> buffer 0
    async_copy_b128(lds_off(&lA[0][r0][c0]), xb + (size_t)(m0 + r0) * TT + c0);
    async_copy_b128(lds_off(&lA[0][r1][c1]), xb + (size_t)(m0 + r1) * TT + c1);
    async_copy_b128(lds_off(&lB[0][r0][c0]), xb + (size_t)(n0 + r0) * TT + c0);
    async_copy_b128(lds_off(&lB[0][r1][c1]), xb + (size_t)(n0 + r1) * TT + c1);

    for (int t0 = 0; t0 < TT; t0 += 32) {
        int cur = (t0 >> 5) & 1;
        wait_async0();          // my DMAs for 'cur' (and any in flight) done
        __syncthreads();        // everyone's DMAs published; prev compute done

        if (t0 + 32 < TT) {
            int nt = t0 + 32, nb = cur ^ 1;
            async_copy_b128(lds_off(&lA[nb][r0][c0]), xb + (size_t)(m0 + r0) * TT + nt + c0);
            async_copy_b128(lds_off(&lA[nb][r1][c1]), xb + (size_t)(m0 + r1) * TT + nt + c1);
            async_copy_b128(lds_off(&lB[nb][r0][c0]), xb + (size_t)(n0 + r0) * TT + nt + c0);
            async_copy_b128(lds_off(&lB[nb][r1][c1]), xb + (size_t)(n0 + r1) * TT + nt + c1);
        }

#pragma unroll
        for (int k = 0; k < 32; k += 4) {
            int kb = k + 2 * (lane >> 4);
            int ml = lane & 15;
            v2f a  = *(const v2f*)&lA[cur][wm * 16 + ml][kb];
            v2f b0 = *(const v2f*)&lB[cur][(wn * 2 + 0) * 16 + ml][kb];
            v2f b1 = *(const v2f*)&lB[cur][(wn * 2 + 1) * 16 + ml][kb];
            acc0 = wmma_f32(a, b0, acc0);
            acc1 = wmma_f32(a, b1, acc1);
        }
    }

    float sc  = scale[b];
    float s2b = s2[b];
    const float* ub = u + (size_t)b * CH;
    int M = (lane >> 4) * 8;
    int N = lane & 15;
    float un0 = ub[n0 + (wn * 2 + 0) * 16 + N];
    float un1 = ub[n0 + (wn * 2 + 1) * 16 + N];
    float* pb = pcc + (size_t)b * CH * CH;
#pragma unroll
    for (int r = 0; r < 8; ++r) {
        float um = ub[m0 + wm * 16 + M + r];
        pb[(size_t)(m0 + wm * 16 + M + r) * CH + n0 + (wn * 2 + 0) * 16 + N] =
            (acc0[r] - um - un0 + s2b) * sc;
        pb[(size_t)(m0 + wm * 16 + M + r) * CH + n0 + (wn * 2 + 1) * 16 + N] =
            (acc1[r] - um - un1 + s2b) * sc;
    }
}

// ---------------------------------------------------------------------------
// Kernel 5: q/k projection + scores + 3-pass softmax -> attn (d_out tail).
// ---------------------------------------------------------------------------
__global__ void k_attn(const float* __restrict__ pcc,
                       const float* __restrict__ q_w,
                       const float* __restrict__ q_b,
                       const float* __restrict__ k_w,
                       const float* __restrict__ k_b,
                       float* __restrict__ attn_out) {
    int b = blockIdx.x;
    int r = threadIdx.x;

    __shared__ float lK[CH][EMB];

    const float* prow = pcc + (size_t)b * CH * CH + (size_t)r * CH;
    float qr[EMB], kr[EMB];
#pragma unroll
    for (int e = 0; e < EMB; ++e) { qr[e] = q_b[e]; kr[e] = k_b[e]; }

    for (int d = 0; d < CH; ++d) {
        float p = prow[d];
#pragma unroll
        for (int e = 0; e < EMB; ++e) {
            qr[e] += p * q_w[e * CH + d];
            kr[e] += p * k_w[e * CH + d];
        }
    }
#pragma unroll
    for (int e = 0; e < EMB; ++e) lK[r][e] = kr[e];
    __syncthreads();

    const float sc = 0.0625f;   // 1/sqrt(256)
    float mx = -3.402823466e38f;
    for (int d = 0; d < CH; ++d) {
        float s = 0.0f;
#pragma unroll
        for (int e = 0; e < EMB; ++e) s += qr[e] * lK[d][e];
        mx = fmaxf(mx, s * sc);
    }
    float sum = 0.0f;
    for (int d = 0; d < CH; ++d) {
        float s = 0.0f;
#pragma unroll
        for (int e = 0; e < EMB; ++e) s += qr[e] * lK[d][e];
        sum += __expf(s * sc - mx);
    }
    float inv = 1.0f / sum;
    float* ao = attn_out + (size_t)b * CH * CH + (size_t)r * CH;
    for (int d = 0; d < CH; ++d) {
        float s = 0.0f;
#pragma unroll
        for (int e = 0; e < EMB; ++e) s += qr[e] * lK[d][e];
        ao[d] = __expf(s * sc - mx) * inv;
    }
}

// ---------------------------------------------------------------------------
// Kernel 6: out = attn @ x via WMMA with async-DMA double buffering.
// Block = 256 threads (8 waves): 64(m) x 128(t) tile, K = 256 in chunks of 16.
// ---------------------------------------------------------------------------
__global__ void k_out(const float* __restrict__ attn,
                      const float* __restrict__ x,
                      float* __restrict__ out) {
    int b  = blockIdx.x;
    int m0 = blockIdx.y * 64;
    int t0 = blockIdx.z * 128;

    __shared__ float lA[2][64][20];    // 80B rows: 16B-aligned, conflict-free cols
    __shared__ float lB[2][16][132];   // 528B rows: 16B-aligned

    int tid  = threadIdx.x;
    int lane = tid & 31;
    int w    = tid >> 5;

    const float* ab = attn + (size_t)b * CH * CH;
    const float* xb = x + (size_t)b * CH * TT;

    // A chunk 64x16: one 16B op/thread. B chunk 16x128: two 16B ops/thread.
    int ar = tid >> 2, ac = (tid & 3) * 4;
    int br0 = (tid * 4) >> 7, bc0 = (tid * 4) & 127;
    int br1 = (tid * 4 + 1024) >> 7, bc1 = (tid * 4 + 1024) & 127;

    v8f acc[4] = {};

    async_copy_b128(lds_off(&lA[0][ar][ac]), ab + (size_t)(m0 + ar) * CH + ac);
    async_copy_b128(lds_off(&lB[0][br0][bc0]), xb + (size_t)br0 * TT + t0 + bc0);
    async_copy_b128(lds_off(&lB[0][br1][bc1]), xb + (size_t)br1 * TT + t0 + bc1);

    for (int d0 = 0; d0 < CH; d0 += 16) {
        int cur = (d0 >> 4) & 1;
        wait_async0();
        __syncthreads();

        if (d0 + 16 < CH) {
            int nd = d0 + 16, nb = cur ^ 1;
            async_copy_b128(lds_off(&lA[nb][ar][ac]), ab + (size_t)(m0 + ar) * CH + nd + ac);
            async_copy_b128(lds_off(&lB[nb][br0][bc0]), xb + (size_t)(nd + br0) * TT + t0 + bc0);
            async_copy_b128(lds_off(&lB[nb][br1][bc1]), xb + (size_t)(nd + br1) * TT + t0 + bc1);
        }

#pragma unroll
        for (int kk = 0; kk < 16; kk += 4) {
            int kb = kk + 2 * (lane >> 4);
            int nl = lane & 15;
            v2f bf;
            bf.x = lB[cur][kb + 0][w * 16 + nl];
            bf.y = lB[cur][kb + 1][w * 16 + nl];
#pragma unroll
            for (int mi = 0; mi < 4; ++mi) {
                v2f a = *(const v2f*)&lA[cur][mi * 16 + nl][kb];
                acc[mi] = wmma_f32(a, bf, acc[mi]);
            }
        }
    }

    int M = (lane >> 4) * 8;
    int N = lane & 15;
#pragma unroll
    for (int mi = 0; mi < 4; ++mi) {
#pragma unroll
        for (int r = 0; r < 8; ++r) {
            out[(size_t)b * CH * TT + (size_t)(m0 + mi * 16 + M + r) * TT
                + t0 + w * 16 + N] = acc[mi][r];
        }
    }
}

// ---------------------------------------------------------------------------
extern "C" void kernel_launch(void* const* d_in, const int* in_sizes, int n_in,
                              void* d_out, int out_size, void* d_ws, size_t ws_size,
                              hipStream_t stream) {
    (void)in_sizes; (void)n_in; (void)out_size; (void)ws_size;

    const float* x   = (const float*)d_in[0];
    const float* q_w = (const float*)d_in[1];
    const float* q_b = (const float*)d_in[2];
    const float* k_w = (const float*)d_in[3];
    const float* k_b = (const float*)d_in[4];

    float* out      = (float*)d_out;                      // [B, C, T]
    float* attn_out = out + (size_t)BATCH * CH * TT;      // [B, C, C]

    // workspace layout (floats)
    float* ws       = (float*)d_ws;
    float* mean     = ws;                                  // B*T   = 131072
    float* partials = mean + (size_t)BATCH * TT;           // 512
    float* scale    = partials + 512;                      // 16
    float* s2       = scale + 16;                          // 16
    float* u        = s2 + 16;                             // B*C   = 4096
    float* pcc      = u + (size_t)BATCH * CH;              // B*C*C = 1048576

    k_mean_ssq<<<dim3((BATCH * TT) / 256), dim3(256), 0, stream>>>(x, mean, partials);
    k_u<<<dim3(BATCH * CH), dim3(256), 0, stream>>>(x, mean, u);
    k_scale<<<dim3(BATCH), dim3(256), 0, stream>>>(partials, mean, scale, s2);
    k_cov<<<dim3(BATCH, CH / 64, CH / 64), dim3(256), 0, stream>>>(x, u, scale, s2, pcc);
    k_attn<<<dim3(BATCH), dim3(CH), 0, stream>>>(pcc, q_w, q_b, k_w, k_b, attn_out);
    k_out<<<dim3(BATCH, CH / 64, TT / 128), dim3(256), 0, stream>>>(attn_out, x, out);
}